// SambaAttention_24704651887232
// MI455X (gfx1250) — compile-verified
//
#include <hip/hip_runtime.h>
#include <hip/hip_bf16.h>
#include <math.h>

// ---- problem constants (match reference) ----
#define HIDDEN     2048
#define N_HEADS    32
#define N_KV       8
#define HEAD_DIM   64
#define BLOCK_SIZE 16
#define BATCH      32
#define MAX_BLOCKS 128
#define MAX_SEQ    (MAX_BLOCKS * BLOCK_SIZE)
#define OP_SIZE    (N_HEADS * HEAD_DIM + 2 * N_KV * HEAD_DIM)  // 3072
#define Q_SIZE     (N_HEADS * HEAD_DIM)                        // 2048
#define KV_OFF     Q_SIZE                                      // 2048
#define V_OFF      (Q_SIZE + N_KV * HEAD_DIM)                  // 2560

typedef __attribute__((ext_vector_type(2))) float v2f;
typedef __attribute__((ext_vector_type(8))) float v8f;

// ---------------------------------------------------------------------------
// Wave-level GEMM: C[32,N] = A[32,K] * B[N,K]^T (row-major, fp32).
// One wave per 16-wide N-tile; each wave computes BOTH 16-row M-tiles so the
// weight (B) fragment is loaded from HBM once and fed to two WMMAs. K-loop in
// steps of 8 = 4 V_WMMA_F32_16X16X4_F32 per iteration across two independent
// accumulator pairs (breaks the WMMA->WMMA RAW hazard chain).
// A-frag: lane l<16 holds A[m][k0+{0,1}], lane l>=16 holds K={2,3}.
// B-frag: lane l<16 holds B[n0+l][k0+{0,1}], lane l>=16 holds K={2,3}.
// ---------------------------------------------------------------------------
__global__ __launch_bounds__(32) void gemm_nt_f32_wmma_m32(
    const float* __restrict__ A, const float* __restrict__ Bw,
    float* __restrict__ C, int N, int K) {
  const int lane = threadIdx.x & 31;
  const int n0 = blockIdx.x << 4;
  const int r  = lane & 15;
  const int kq = (lane >> 4) << 1;  // 0 or 2

  const float* __restrict__ ap0 = A  + (size_t)r * K + kq;         // rows 0..15
  const float* __restrict__ ap1 = A  + (size_t)(16 + r) * K + kq;  // rows 16..31
  const float* __restrict__ bp  = Bw + (size_t)(n0 + r) * K + kq;

  v8f acc00 = {}, acc01 = {}, acc10 = {}, acc11 = {};
  for (int k0 = 0; k0 < K; k0 += 8) {
    v2f bl  = *(const v2f*)(bp  + k0);
    v2f a0l = *(const v2f*)(ap0 + k0);
    v2f a1l = *(const v2f*)(ap1 + k0);
    v2f bh  = *(const v2f*)(bp  + k0 + 4);
    v2f a0h = *(const v2f*)(ap0 + k0 + 4);
    v2f a1h = *(const v2f*)(ap1 + k0 + 4);
    acc00 = __builtin_amdgcn_wmma_f32_16x16x4_f32(false, a0l, false, bl, (short)0, acc00, false, false);
    acc10 = __builtin_amdgcn_wmma_f32_16x16x4_f32(false, a1l, false, bl, (short)0, acc10, false, false);
    acc01 = __builtin_amdgcn_wmma_f32_16x16x4_f32(false, a0h, false, bh, (short)0, acc01, false, false);
    acc11 = __builtin_amdgcn_wmma_f32_16x16x4_f32(false, a1h, false, bh, (short)0, acc11, false, false);
  }
  v8f d0 = acc00 + acc01;
  v8f d1 = acc10 + acc11;

  // D layout: lane l holds col n=l%16, VGPR i holds row m = i + 8*(l/16)
  const int mb = (lane >> 4) << 3;
  float* cp0 = C + (size_t)mb * N + n0 + r;
  float* cp1 = C + (size_t)(16 + mb) * N + n0 + r;
#pragma unroll
  for (int i = 0; i < 8; ++i) {
    cp0[(size_t)i * N] = d0[i];
    cp1[(size_t)i * N] = d1[i];
  }
}

// ---------------------------------------------------------------------------
// RoPE applied in place to q (32 heads) and k (8 heads) in the qkv buffer.
// One thread per (batch, head, d<32). q heads are at head*64, k heads start
// at offset 2048 == 32*64, so base = b*OP_SIZE + head*64 for head in [0,40).
// ---------------------------------------------------------------------------
__global__ void rope_kernel(float* __restrict__ qkv,
                            const int* __restrict__ positions) {
  const int HALF = HEAD_DIM / 2;            // 32
  const int NH = N_HEADS + N_KV;            // 40
  const int idx = blockIdx.x * blockDim.x + threadIdx.x;
  if (idx >= BATCH * NH * HALF) return;
  const int d    = idx % HALF;
  const int head = (idx / HALF) % NH;
  const int b    = idx / (HALF * NH);

  const float pos = (float)positions[b];
  // inv_freq = 10000^{-d/32} = exp(-d * ln(10000)/32)
  const float inv = __expf(-(float)d * 0.2878231366f);
  float sn, cs;
  __sincosf(pos * inv, &sn, &cs);

  const size_t base = (size_t)b * OP_SIZE + (size_t)head * HEAD_DIM;
  const float x1 = qkv[base + d];
  const float x2 = qkv[base + d + HALF];
  qkv[base + d]        = x1 * cs - x2 * sn;
  qkv[base + d + HALF] = x2 * cs + x1 * sn;
}

// ---------------------------------------------------------------------------
// Flash-decode paged attention. Grid (B, N_KV); 256 threads = 8 waves.
// Wave w handles seq positions s = w, w+8, ... Each lane owns dims {2l,2l+1}
// of the 64-dim head. Scores for the 4 grouped q-heads are produced by a
// wave32 butterfly reduction; online softmax keeps m/l wave-uniform.
// Position seq_len-1 reads the freshly projected (RoPE'd) k/v instead of the
// cache, matching the reference's scatter-then-attend semantics.
// ---------------------------------------------------------------------------
__global__ __launch_bounds__(256) void attn_decode_kernel(
    const float* __restrict__ qkv, const float* __restrict__ kcache,
    const float* __restrict__ vcache, const int* __restrict__ btab,
    const int* __restrict__ seq_lens, float* __restrict__ attn_out) {
  const int b    = blockIdx.x;
  const int kv   = blockIdx.y;
  const int tid  = threadIdx.x;
  const int wave = tid >> 5;
  const int lane = tid & 31;

  const int seqlen = seq_lens[b];
  const int last   = seqlen - 1;
  const float scale = 0.125f;  // 64^-0.5

  // q fragments: 4 grouped heads, 2 dims per lane
  float2 qf[4];
#pragma unroll
  for (int h = 0; h < 4; ++h)
    qf[h] = *(const float2*)(qkv + (size_t)b * OP_SIZE +
                             (size_t)(kv * 4 + h) * HEAD_DIM + 2 * lane);

  float m[4]  = {-1e30f, -1e30f, -1e30f, -1e30f};
  float l[4]  = {0.f, 0.f, 0.f, 0.f};
  float ax[4] = {0.f, 0.f, 0.f, 0.f};
  float ay[4] = {0.f, 0.f, 0.f, 0.f};

  const int* bt = btab + (size_t)b * MAX_BLOCKS;
  const float* newk = qkv + (size_t)b * OP_SIZE + KV_OFF + (size_t)kv * HEAD_DIM;
  const float* newv = qkv + (size_t)b * OP_SIZE + V_OFF  + (size_t)kv * HEAD_DIM;

  for (int s = wave; s < seqlen; s += 8) {
    const float *kp, *vp;
    if (s == last) {
      kp = newk; vp = newv;
    } else {
      const int blk = bt[s >> 4];
      const size_t base =
          ((((size_t)blk << 4) + (s & 15)) * N_KV + kv) * HEAD_DIM;
      kp = kcache + base;
      vp = vcache + base;
    }
    // prefetch next K line (global_prefetch_b8) to hide table indirection
    const int s2 = s + 8;
    if (s2 < seqlen && s2 != last) {
      const int blk2 = bt[s2 >> 4];
      const float* kp2 =
          kcache + ((((size_t)blk2 << 4) + (s2 & 15)) * N_KV + kv) * HEAD_DIM;
      __builtin_prefetch(kp2 + 2 * lane, 0, 1);
    }

    const float2 kx = *(const float2*)(kp + 2 * lane);
    float p[4];
#pragma unroll
    for (int h = 0; h < 4; ++h)
      p[h] = qf[h].x * kx.x + qf[h].y * kx.y;
    // wave32 butterfly sum -> full dot product in every lane
#pragma unroll
    for (int off = 16; off > 0; off >>= 1) {
#pragma unroll
      for (int h = 0; h < 4; ++h) p[h] += __shfl_xor(p[h], off, 32);
    }

    const float2 vx = *(const float2*)(vp + 2 * lane);
#pragma unroll
    for (int h = 0; h < 4; ++h) {
      const float sc   = p[h] * scale;
      const float mn   = fmaxf(m[h], sc);
      const float corr = __expf(m[h] - mn);
      const float w    = __expf(sc - mn);
      l[h]  = l[h] * corr + w;
      ax[h] = ax[h] * corr + w * vx.x;
      ay[h] = ay[h] * corr + w * vx.y;
      m[h]  = mn;
    }
  }

  // cross-wave combine in LDS
  __shared__ float sm[8][4];
  __shared__ float sl[8][4];
  __shared__ float sacc[8][4][HEAD_DIM];
#pragma unroll
  for (int h = 0; h < 4; ++h) {
    if (lane == 0) { sm[wave][h] = m[h]; sl[wave][h] = l[h]; }
    sacc[wave][h][2 * lane]     = ax[h];
    sacc[wave][h][2 * lane + 1] = ay[h];
  }
  __syncthreads();

  const int h = tid >> 6;   // 0..3
  const int d = tid & 63;   // 0..63
  float M = -1e30f;
#pragma unroll
  for (int w = 0; w < 8; ++w) M = fmaxf(M, sm[w][h]);
  float L = 0.f, O = 0.f;
#pragma unroll
  for (int w = 0; w < 8; ++w) {
    const float c = __expf(sm[w][h] - M);
    L += sl[w][h] * c;
    O += sacc[w][h][d] * c;
  }
  attn_out[(size_t)b * Q_SIZE + (size_t)(kv * 4 + h) * HEAD_DIM + d] = O / L;
}

// ---------------------------------------------------------------------------
extern "C" void kernel_launch(void* const* d_in, const int* in_sizes, int n_in,
                              void* d_out, int out_size, void* d_ws,
                              size_t ws_size, hipStream_t stream) {
  const float* hidden    = (const float*)d_in[0];
  const float* Wqkv      = (const float*)d_in[1];  // [OP_SIZE, HIDDEN]
  const float* Wout      = (const float*)d_in[2];  // [HIDDEN, HIDDEN]
  const float* kcache    = (const float*)d_in[3];
  const float* vcache    = (const float*)d_in[4];
  const int*   btab      = (const int*)d_in[5];
  const int*   seq_lens  = (const int*)d_in[6];
  const int*   positions = (const int*)d_in[7];
  float* out = (float*)d_out;

  float* qkv  = (float*)d_ws;                       // B*OP_SIZE floats
  float* attn = qkv + (size_t)BATCH * OP_SIZE;      // B*Q_SIZE floats

  // 1) qkv = hidden @ Wqkv^T   (32 x 3072, K=2048) — weights read once
  gemm_nt_f32_wmma_m32<<<OP_SIZE / 16, 32, 0, stream>>>(hidden, Wqkv, qkv,
                                                        OP_SIZE, HIDDEN);
  // 2) RoPE in place on q and k
  {
    const int total = BATCH * (N_HEADS + N_KV) * (HEAD_DIM / 2);
    rope_kernel<<<(total + 255) / 256, 256, 0, stream>>>(qkv, positions);
  }
  // 3) paged flash-decode attention (streams each K/V line exactly once)
  {
    dim3 g(BATCH, N_KV);
    attn_decode_kernel<<<g, 256, 0, stream>>>(qkv, kcache, vcache, btab,
                                              seq_lens, attn);
  }
  // 4) out = attn @ out_proj^T  (32 x 2048, K=2048) — weights read once
  gemm_nt_f32_wmma_m32<<<HIDDEN / 16, 32, 0, stream>>>(attn, Wout, out,
                                                       HIDDEN, HIDDEN);
}